// TensorQuantizer_57750130262234
// MI455X (gfx1250) — compile-verified
//
#include <hip/hip_runtime.h>

// MX-FP8 (e4m3, block=32, shared e8m0 exponent) fake quantization.
// Bandwidth-bound streaming kernel: 512 MiB traffic, ~23 us floor @ 23.3 TB/s.
// gfx1250 paths: hardware e4m3 converters (v_cvt_pk_fp8_f32 / v_cvt_pk_f32_fp8),
// wave32 DS-permute amax reduction, v_ldexp_f32 exact 2^e scaling, v_med3_f32
// saturation, non-temporal (TH=NT) 128-bit loads/stores, 32-bit addressing.

typedef __attribute__((ext_vector_type(4))) float v4f;
typedef __attribute__((ext_vector_type(2))) float v2f;

#if __has_builtin(__builtin_amdgcn_cvt_pk_fp8_f32) && __has_builtin(__builtin_amdgcn_cvt_pk_f32_fp8)
#define USE_HW_FP8 1
#else
#define USE_HW_FP8 0
#endif

// Software e4m3 quant-dequant fallback (RNE onto grid, input pre-clamped).
__device__ __forceinline__ float quant_e4m3_sw(float c) {
    float ax = __builtin_fabsf(c);
    int pe = (int)((__float_as_uint(ax) >> 23) & 0xFF) - 127; // denorm/zero -> -127
    pe = pe < -6 ? -6 : pe;
    float q = __builtin_rintf(__builtin_ldexpf(c, 3 - pe));   // v_rndne_f32
    return __builtin_ldexpf(q, pe - 3);
}

__global__ __launch_bounds__(256) void mxfp8_fakequant_kernel(
    const float* __restrict__ in, float* __restrict__ out, unsigned n) {
    unsigned t    = blockIdx.x * 256u + threadIdx.x;
    unsigned base = t * 8u;                 // 32-bit indexing: n = 64Mi < 2^31
    if (base + 8u > n) return;              // n is a multiple of 32 here

    // 8 contiguous elements per lane; one MX block (32) = 4 adjacent lanes.
    const v4f* ip = (const v4f*)(in + base);
    v4f a = __builtin_nontemporal_load(ip);
    v4f b = __builtin_nontemporal_load(ip + 1);

    // In-lane amax over 8 values (compiler fuses into v_max3_num_f32 |..|).
    float m = fmaxf(fmaxf(fmaxf(__builtin_fabsf(a.x), __builtin_fabsf(a.y)),
                          fmaxf(__builtin_fabsf(a.z), __builtin_fabsf(a.w))),
                    fmaxf(fmaxf(__builtin_fabsf(b.x), __builtin_fabsf(b.y)),
                          fmaxf(__builtin_fabsf(b.z), __builtin_fabsf(b.w))));
    // Cross-lane amax within the 4-lane MX block (wave32 DS-permute).
    m = fmaxf(m, __shfl_xor(m, 1, 32));
    m = fmaxf(m, __shfl_xor(m, 2, 32));

    // shared_exp = clip(floor(log2(amax or TINY)) - 8, -127, 127)
    // floor(log2) via exponent field; renormalize by 2^24 for fp32 denorms;
    // amax == 0 uses TINY -> floor(log2) = -126.
    bool  sub = m < 0x1p-126f;
    float mn  = sub ? m * 0x1p24f : m;
    int e = (int)((__float_as_uint(mn) >> 23) & 0xFF) - 127 - (sub ? 24 : 0);
    if (m == 0.0f) e = -126;
    e -= 8;                                 // EMAX for e4m3
    e = e < -127 ? -127 : (e > 127 ? 127 : e);

    // Scale into element range (exact v_ldexp_f32); saturate with v_med3_f32.
    float s[8] = {a.x, a.y, a.z, a.w, b.x, b.y, b.z, b.w};
#pragma unroll
    for (int i = 0; i < 8; ++i) {
        float xs = __builtin_ldexpf(s[i], -e);
        s[i] = __builtin_amdgcn_fmed3f(xs, -448.0f, 448.0f);
    }

#if USE_HW_FP8
    // Hardware e4m3 RNE quant-dequant: v_cvt_pk_fp8_f32 + v_cvt_pk_f32_fp8.
    int p01 = __builtin_amdgcn_cvt_pk_fp8_f32(s[0], s[1], 0, false);
    int p23 = __builtin_amdgcn_cvt_pk_fp8_f32(s[2], s[3], 0, false);
    int p45 = __builtin_amdgcn_cvt_pk_fp8_f32(s[4], s[5], 0, false);
    int p67 = __builtin_amdgcn_cvt_pk_fp8_f32(s[6], s[7], 0, false);
    v2f d01 = __builtin_amdgcn_cvt_pk_f32_fp8(p01, false);
    v2f d23 = __builtin_amdgcn_cvt_pk_f32_fp8(p23, false);
    v2f d45 = __builtin_amdgcn_cvt_pk_f32_fp8(p45, false);
    v2f d67 = __builtin_amdgcn_cvt_pk_f32_fp8(p67, false);
    s[0] = d01.x; s[1] = d01.y; s[2] = d23.x; s[3] = d23.y;
    s[4] = d45.x; s[5] = d45.y; s[6] = d67.x; s[7] = d67.y;
#else
#pragma unroll
    for (int i = 0; i < 8; ++i) s[i] = quant_e4m3_sw(s[i]);
#endif

    // Rescale (exact) and stream out with NT hint.
    v4f r0, r1;
    r0.x = __builtin_ldexpf(s[0], e);
    r0.y = __builtin_ldexpf(s[1], e);
    r0.z = __builtin_ldexpf(s[2], e);
    r0.w = __builtin_ldexpf(s[3], e);
    r1.x = __builtin_ldexpf(s[4], e);
    r1.y = __builtin_ldexpf(s[5], e);
    r1.z = __builtin_ldexpf(s[6], e);
    r1.w = __builtin_ldexpf(s[7], e);

    v4f* op = (v4f*)(out + base);
    __builtin_nontemporal_store(r0, op);
    __builtin_nontemporal_store(r1, op + 1);
}

extern "C" void kernel_launch(void* const* d_in, const int* in_sizes, int n_in,
                              void* d_out, int out_size, void* d_ws, size_t ws_size,
                              hipStream_t stream) {
    const float* in = (const float*)d_in[0];
    float* out = (float*)d_out;
    unsigned n = (unsigned)in_sizes[0];      // 8192*8192, multiple of 32
    unsigned threads = (n + 7u) / 8u;
    unsigned block = 256u;
    unsigned grid = (threads + block - 1u) / block;
    mxfp8_fakequant_kernel<<<dim3(grid), dim3(block), 0, stream>>>(in, out, n);
}